// HistogramLoss_87385404605092
// MI455X (gfx1250) — compile-verified
//
#include <hip/hip_runtime.h>

// ---------------------------------------------------------------------------
// Soft-histogram Hellinger loss for MI455X (gfx1250, wave32).
//   sigma = 0.5/64, bin spacing = 1/63  =>  R = spacing/sigma = 2.0317
//   w_k = exp(-0.5*R^2*(d-k)^2) = E0 * r^k * G_k,  E0=exp(-0.5R^2 d^2), r=exp(R^2 d)
//   Only |k|<=3 contributes above f32 noise (G4 ~ 1e-15).
// Kernel 1: 48 channels x 16 slices, per-wave LDS hists (ds_add_f32),
//           cross-wave reduce with v_wmma_f32_16x16x4_f32 (ones x H).
// Kernel 2: per-channel Hellinger.  Kernel 3: mean over 48 channels.
// ---------------------------------------------------------------------------

#define NBINS   64
#define SLICES  16
#define PAD     3
#define WSTRIDE 72          // 3 guard + 64 bins + 3 guard + 2 pad
#define EPSF    1e-8f

typedef __attribute__((ext_vector_type(2))) float v2f;
typedef __attribute__((ext_vector_type(8))) float v8f;

__device__ __forceinline__ float clampf(float v, float lo, float hi) {
    return fminf(fmaxf(v, lo), hi);
}

__global__ __launch_bounds__(256) void hist_partial_kernel(
    const float* __restrict__ real, const float* __restrict__ gen,
    float* __restrict__ partial)   // [48][SLICES][2][64]
{
    __shared__ float wh[8][2][WSTRIDE];

    const int tid   = threadIdx.x;
    const int wave  = tid >> 5;
    const int lane  = tid & 31;
    const int blk   = blockIdx.x;
    const int ch    = blk >> 4;        // 0..47  (b*3 + c)
    const int slice = blk & 15;        // 0..15

    for (int i = tid; i < 8 * 2 * WSTRIDE; i += 256) ((float*)wh)[i] = 0.0f;
    __syncthreads();

    const float R2 = 4.1279922f;                 // (128/63)^2
    const float G1 = 0.126918f;                  // exp(-0.5*R2)
    const float G2 = 2.59700e-4f;                // exp(-2.0*R2)
    const float G3 = 8.5620e-9f;                 // exp(-4.5*R2)

    const long base4 = (long)ch * 16384 + (long)slice * 1024;  // in float4 units

    for (int img = 0; img < 2; ++img) {
        const float4* __restrict__ src =
            (const float4*)(img == 0 ? real : gen) + base4;
        float* h = &wh[wave][img][PAD];
        #pragma unroll
        for (int i = 0; i < 4; ++i) {
            if (i < 3) __builtin_prefetch(&src[tid + (i + 1) * 256], 0, 3);
            float4 p4 = src[tid + i * 256];
            float px[4] = {p4.x, p4.y, p4.z, p4.w};
            #pragma unroll
            for (int j = 0; j < 4; ++j) {
                float x   = fmaf(px[j], 0.5f, 0.5f);          // (p+1)/2
                float u   = x * 63.0f;                        // in bin units
                float jcf = clampf(rintf(u), 0.0f, 63.0f);
                int   jc  = (int)jcf;
                float d   = clampf(u - jcf, -8.0f, 8.0f);     // |d|>8 => w==0
                float E0  = __expf(-0.5f * R2 * d * d);
                float r   = __expf(R2 * d);
                float rm  = __builtin_amdgcn_rcpf(r);
                float r2  = r * r,  r3  = r2 * r;
                float rm2 = rm * rm, rm3 = rm2 * rm;
                float e1 = E0 * G1, e2 = E0 * G2, e3 = E0 * G3;
                float* hb = h + jc;                 // guards absorb edge k
                atomicAdd(hb - 3, e3 * rm3);
                atomicAdd(hb - 2, e2 * rm2);
                atomicAdd(hb - 1, e1 * rm);
                atomicAdd(hb + 0, E0);
                atomicAdd(hb + 1, e1 * r);
                atomicAdd(hb + 2, e2 * r2);
                atomicAdd(hb + 3, e3 * r3);
            }
        }
    }
    __syncthreads();

    // Cross-wave reduction with f32 WMMA:
    //   out[16 bins] = ones(16x4) x H(4x16)  chained twice (8 wave hists).
    // 8 groups = 2 images x 4 bin-blocks; wave g handles group g.
    {
        const int imgi = wave >> 2;
        const int bb   = wave & 3;
        const int n    = lane & 15;
        const int kw   = (lane >> 4) << 1;   // 0 or 2 (K rows held by this lane)

        v2f a;  a.x = 1.0f; a.y = 1.0f;      // A = all-ones 16x4
        v2f b0, b1;
        b0.x = wh[kw + 0][imgi][PAD + bb * 16 + n];
        b0.y = wh[kw + 1][imgi][PAD + bb * 16 + n];
        b1.x = wh[kw + 4][imgi][PAD + bb * 16 + n];
        b1.y = wh[kw + 5][imgi][PAD + bb * 16 + n];

        v8f acc = {};
        acc = __builtin_amdgcn_wmma_f32_16x16x4_f32(
                  false, a, false, b0, (short)0, acc, false, false);
        acc = __builtin_amdgcn_wmma_f32_16x16x4_f32(
                  false, a, false, b1, (short)0, acc, false, false);

        if (lane < 16) {    // D row M=0 lives in VGPR0, lanes 0..15 = N
            partial[(((long)ch * SLICES + slice) * 2 + imgi) * NBINS + bb * 16 + n]
                = acc[0];
        }
    }
}

__global__ __launch_bounds__(64) void hellinger_kernel(
    const float* __restrict__ partial, float* __restrict__ hell)
{
    __shared__ float red[64];
    __shared__ float sums[2];
    const int bin = threadIdx.x;     // one bin per thread
    const int ch  = blockIdx.x;

    float h[2];
    for (int img = 0; img < 2; ++img) {
        float s = 0.0f;
        for (int sl = 0; sl < SLICES; ++sl)
            s += partial[(((long)ch * SLICES + sl) * 2 + img) * NBINS + bin];
        h[img] = s;
        red[bin] = s;
        __syncthreads();
        for (int off = 32; off > 0; off >>= 1) {
            if (bin < off) red[bin] += red[bin + off];
            __syncthreads();
        }
        if (bin == 0) sums[img] = red[0];
        __syncthreads();
    }

    float qr = sqrtf(h[0] / (sums[0] + EPSF));
    float qg = sqrtf(h[1] / (sums[1] + EPSF));
    float dq = qr - qg;
    red[bin] = dq * dq;
    __syncthreads();
    for (int off = 32; off > 0; off >>= 1) {
        if (bin < off) red[bin] += red[bin + off];
        __syncthreads();
    }
    if (bin == 0) hell[ch] = sqrtf(0.5f * red[0]);   // (1/sqrt2)*||.||_2
}

__global__ __launch_bounds__(64) void mean_kernel(
    const float* __restrict__ hell, float* __restrict__ out)
{
    __shared__ float red[64];
    const int t = threadIdx.x;
    red[t] = (t < 48) ? hell[t] : 0.0f;
    __syncthreads();
    for (int off = 32; off > 0; off >>= 1) {
        if (t < off) red[t] += red[t + off];
        __syncthreads();
    }
    if (t == 0) out[0] = red[0] * (1.0f / 48.0f);
}

extern "C" void kernel_launch(void* const* d_in, const int* in_sizes, int n_in,
                              void* d_out, int out_size, void* d_ws, size_t ws_size,
                              hipStream_t stream) {
    const float* real = (const float*)d_in[0];
    const float* gen  = (const float*)d_in[1];
    // d_in[2] (bin_centers) is linspace(0,1,64): constants are baked in.

    float* ws      = (float*)d_ws;
    float* partial = ws;                               // 48*16*2*64 = 98304 f32
    float* hell    = ws + 48 * SLICES * 2 * NBINS;     // +48 f32

    hist_partial_kernel<<<48 * SLICES, 256, 0, stream>>>(real, gen, partial);
    hellinger_kernel<<<48, 64, 0, stream>>>(partial, hell);
    mean_kernel<<<1, 64, 0, stream>>>(hell, (float*)d_out);
}